// NormalizedBoxCenterEncoder_36567351558162
// MI455X (gfx1250) — compile-verified
//
#include <hip/hip_runtime.h>

typedef float v4f __attribute__((ext_vector_type(4)));

#define BDIM 256
#define ITER 8
#define M_REF 500

__global__ __launch_bounds__(BDIM) void box_encode_kernel(
    const float* __restrict__ samples,
    const int*   __restrict__ matches,
    const v4f*   __restrict__ anchors,
    const v4f*   __restrict__ refs,
    v4f*         __restrict__ out_tgt,
    v4f*         __restrict__ out_msk,
    int N)
{
    __shared__ v4f sref[M_REF];   // 8 KB: per-batch reference-box table

    const int tid = threadIdx.x;
    const int b   = blockIdx.y;

    // ---- Stage refs[b] (500 boxes = 8 KB) into LDS via CDNA5 async copy ----
    // GVS addressing: mem = SGPR_base + VGPR_offset; VDST VGPR = LDS byte addr.
    {
        unsigned long long gbase =
            (unsigned long long)(const void*)(refs + (size_t)b * M_REF);
        for (int idx = tid; idx < M_REF; idx += BDIM) {
            unsigned lds_addr = (unsigned)(unsigned long long)(const void*)&sref[idx];
            unsigned goff     = (unsigned)idx * 16u;
            asm volatile("global_load_async_to_lds_b128 %0, %1, %2"
                         :: "v"(lds_addr), "v"(goff), "s"(gbase)
                         : "memory");
        }
        asm volatile("s_wait_asynccnt 0" ::: "memory");
    }
    __syncthreads();

    // Corner -> center form in place (done once per block, reused ~4x per box)
    for (int idx = tid; idx < M_REF; idx += BDIM) {
        v4f c = sref[idx];
        float w = c.z - c.x;
        float h = c.w - c.y;
        v4f g;
        g.x = c.x + 0.5f * w;
        g.y = c.y + 0.5f * h;
        g.z = w;
        g.w = h;
        sref[idx] = g;
    }
    __syncthreads();

    const size_t bn = (size_t)b * N;
    const float* sb = samples + bn;
    const int*   mb = matches + bn;
    const v4f*   ab = anchors + bn;
    v4f*         tb = out_tgt + bn;
    v4f*         kb = out_msk + bn;

    const float LN2x5 = 3.4657359028f;  // ln(2) * 5   (STD 0.2 -> 1/0.2 = 5)
    int n = blockIdx.x * (BDIM * ITER) + tid;
#pragma unroll
    for (int it = 0; it < ITER; ++it, n += BDIM) {
        if (n < N) {
            // Streaming reads, touched once -> nontemporal hints
            float s = __builtin_nontemporal_load(sb + n);
            int   m = __builtin_nontemporal_load(mb + n);
            v4f   a = __builtin_nontemporal_load(ab + n);

            m = m < 0 ? 0 : (m >= M_REF ? M_REF - 1 : m);
            v4f g = sref[m];               // ds_load_b128 gather from LDS

            float aw  = a.z - a.x;
            float ah  = a.w - a.y;
            float rw  = __builtin_amdgcn_rcpf(aw);   // v_rcp_f32
            float rh  = __builtin_amdgcn_rcpf(ah);
            float acx = a.x + 0.5f * aw;
            float acy = a.y + 0.5f * ah;

            v4f t;
            t.x = (g.x - acx) * rw * 10.0f;                 // STD 0.1 -> x10
            t.y = (g.y - acy) * rh * 10.0f;
            t.z = __builtin_amdgcn_logf(g.z * rw) * LN2x5;  // v_log_f32, ln = log2*ln2
            t.w = __builtin_amdgcn_logf(g.w * rh) * LN2x5;

            bool pos = s > 0.5f;
            v4f zero = (v4f){0.f, 0.f, 0.f, 0.f};
            v4f one  = (v4f){1.f, 1.f, 1.f, 1.f};
            v4f tgt = pos ? t : zero;
            v4f msk = pos ? one : zero;

            __builtin_nontemporal_store(tgt, tb + n);   // b128 NT streaming stores
            __builtin_nontemporal_store(msk, kb + n);
        }
    }
}

extern "C" void kernel_launch(void* const* d_in, const int* in_sizes, int n_in,
                              void* d_out, int out_size, void* d_ws, size_t ws_size,
                              hipStream_t stream) {
    const float* samples = (const float*)d_in[0];
    const int*   matches = (const int*)d_in[1];
    const v4f*   anchors = (const v4f*)d_in[2];
    const v4f*   refs    = (const v4f*)d_in[3];

    const int BN = in_sizes[0];                 // B*N   (samples is (B,N))
    const int B  = in_sizes[3] / (M_REF * 4);   // refs is (B,M,4) floats
    const int N  = BN / B;

    v4f* out_tgt = (v4f*)d_out;                 // targets: B*N float4
    v4f* out_msk = out_tgt + (size_t)BN;        // masks follow, B*N float4

    dim3 grid((N + BDIM * ITER - 1) / (BDIM * ITER), B);
    box_encode_kernel<<<grid, dim3(BDIM), 0, stream>>>(
        samples, matches, anchors, refs, out_tgt, out_msk, N);
}